// QuantumLayer_13580686590214
// MI455X (gfx1250) — compile-verified
//
#include <hip/hip_runtime.h>
#include <hip/hip_bf16.h>

typedef __attribute__((ext_vector_type(2))) float v2f;
typedef __attribute__((ext_vector_type(8))) float v8f;

#define NQ 10
#define DIM 1024
#define WPB 8   // waves (samples) per block

// ---------------------------------------------------------------------------
// Prologue: build the 40 real 4x4 gate matrices (complex 2x2 expanded) from
// the 80 weights. Gate order: L0: rot q0..q9 (0..9), cry e0..e9 (10..19),
// L1: rot (20..29), cry (30..39). 16 floats each -> 640 floats in d_ws.
// ---------------------------------------------------------------------------
__global__ void build_gates_kernel(const float* __restrict__ W,
                                   float* __restrict__ G) {
  int g = threadIdx.x;
  if (g >= 40) return;
  int L = g / 20, r = g % 20;
  float u00r, u00i, u01r, u01i, u10r, u10i, u11r, u11i;
  if (r < 10) {
    int base = L * 40 + 3 * r;
    float phi = W[base], th = W[base + 1], om = W[base + 2];
    float ct = cosf(0.5f * th), st = sinf(0.5f * th);
    float ap = 0.5f * (phi + om), am = 0.5f * (phi - om);
    float cap = cosf(ap), sap = sinf(ap);
    float cam = cosf(am), sam = sinf(am);
    u00r =  ct * cap; u00i = -ct * sap;   // c * e^{-i(phi+om)/2}
    u01r = -st * cam; u01i = -st * sam;   // -s * e^{+i(phi-om)/2}
    u10r =  st * cam; u10i = -st * sam;   // s * e^{-i(phi-om)/2}
    u11r =  ct * cap; u11i =  ct * sap;   // c * e^{+i(phi+om)/2}
  } else {
    int e = r - 10;
    float th = W[L * 40 + 30 + e];        // FIXED_W is all zeros
    float ct = cosf(0.5f * th), st = sinf(0.5f * th);
    u00r = ct;  u00i = 0.f; u01r = -st; u01i = 0.f;
    u10r = st;  u10i = 0.f; u11r = ct;  u11i = 0.f;
  }
  // real 4x4 acting on [re0, im0, re1, im1]
  float* M = G + g * 16;
  M[ 0] = u00r; M[ 1] = -u00i; M[ 2] = u01r; M[ 3] = -u01i;
  M[ 4] = u00i; M[ 5] =  u00r; M[ 6] = u01i; M[ 7] =  u01r;
  M[ 8] = u10r; M[ 9] = -u10i; M[10] = u11r; M[11] = -u11i;
  M[12] = u10i; M[13] =  u10r; M[14] = u11i; M[15] =  u11r;
}

// ---------------------------------------------------------------------------
// Main kernel: one wave32 = one sample; statevector interleaved (re,im) in
// LDS; every gate application is a chain of V_WMMA_F32_16X16X4_F32 ops
// (A = replicated real 4x4 gate, B = 16 amplitude pairs, D rows 0-3 / 8-11
// hold the result). launch_bounds(256,8) caps VGPRs (<256 -> no
// s_set_vgpr_msb) and keeps occupancy at the LDS-limited 10 waves/SIMD.
// ---------------------------------------------------------------------------
__global__ __launch_bounds__(256, 8) void qsim_kernel(
    const float* __restrict__ inp, const float* __restrict__ G,
    float* __restrict__ out) {
  __shared__ float2 st[WPB][DIM];   // interleaved (re, im): 64 KB
  const int t = threadIdx.x & 31;   // lane
  const int w = threadIdx.x >> 5;   // wave in block
  const int samp = blockIdx.x * WPB + w;
  float2* sv = st[w];

  // ---- state preparation: kron_q (cos(a_q/2), sin(a_q/2)), all real ----
  float cc[NQ], ss[NQ];
  const float* x = inp + samp * NQ;
  #pragma unroll
  for (int q = 0; q < NQ; ++q) {
    float xv = fminf(fmaxf(x[q], 0.f), 1.f);
    float a = 0.5f * 3.14159265358979323846f * xv;   // (pi*x)/2
    cc[q] = cosf(a);
    ss[q] = sinf(a);
  }
  // lane bits j=0..4 are index bits 0..4  -> qubits 9..5
  float plo = 1.f;
  #pragma unroll
  for (int j = 0; j < 5; ++j)
    plo *= ((t >> j) & 1) ? ss[9 - j] : cc[9 - j];
  #pragma unroll 4
  for (int r = 0; r < 32; ++r) {
    // r bits j=0..4 are index bits 5..9 -> qubits 4..0
    float phi = 1.f;
    #pragma unroll
    for (int j = 0; j < 5; ++j)
      phi *= ((r >> j) & 1) ? ss[4 - j] : cc[4 - j];
    sv[r * 32 + t] = make_float2(phi * plo, 0.f);
  }
  __builtin_amdgcn_wave_barrier();

  // A-operand layout (16x4 f32): lanes 0-15 -> K=0/1, lanes 16-31 -> K=2/3
  const int row4 = (t & 3) * 4;
  const int c0 = (t < 16) ? 0 : 2;
  const int pl = t & 15;
  const bool lo = (t < 16);

  for (int L = 0; L < 2; ++L) {
    // ---- 10 single-qubit Rot gates: 512 pairs = 32 WMMAs each ----
    for (int q = 0; q < NQ; ++q) {
      const float* M = G + (L * 20 + q) * 16;
      v2f a;
      a.x = M[row4 + c0];
      a.y = M[row4 + c0 + 1];
      const int b = 9 - q;
      const int S = 1 << b;
      #pragma unroll 4
      for (int blk = 0; blk < 32; ++blk) {
        int p = blk * 16 + pl;
        int i0 = ((p >> b) << (b + 1)) | (p & (S - 1));
        int idx = lo ? i0 : (i0 | S);
        float2 amp = sv[idx];
        v2f bb;
        bb.x = amp.x;
        bb.y = amp.y;
        v8f cz = {};
        v8f d = __builtin_amdgcn_wmma_f32_16x16x4_f32(
            false, a, false, bb, (short)0, cz, false, false);
        sv[idx] = make_float2(lo ? d[0] : d[2], lo ? d[1] : d[3]);
      }
      __builtin_amdgcn_wave_barrier();
    }
    // ---- 10 CRY ring gates: 256 pairs (control bit = 1) = 16 WMMAs each ----
    for (int e = 0; e < NQ; ++e) {
      const float* M = G + (L * 20 + 10 + e) * 16;
      v2f a;
      a.x = M[row4 + c0];
      a.y = M[row4 + c0 + 1];
      const int u = e, v = (e + 1) % NQ;
      const int bu = 9 - u, bv = 9 - v;
      const int blo = bu < bv ? bu : bv;
      const int bhi = bu < bv ? bv : bu;
      #pragma unroll 4
      for (int blk = 0; blk < 16; ++blk) {
        int p = blk * 16 + pl;
        int x1 = ((p >> blo) << (blo + 1)) | (p & ((1 << blo) - 1));
        int x2 = ((x1 >> bhi) << (bhi + 1)) | (x1 & ((1 << bhi) - 1));
        int i0 = x2 | (1 << bu);               // control bit set
        int idx = lo ? i0 : (i0 | (1 << bv));  // target bit 0 / 1
        float2 amp = sv[idx];
        v2f bb;
        bb.x = amp.x;
        bb.y = amp.y;
        v8f cz = {};
        v8f d = __builtin_amdgcn_wmma_f32_16x16x4_f32(
            false, a, false, bb, (short)0, cz, false, false);
        sv[idx] = make_float2(lo ? d[0] : d[2], lo ? d[1] : d[3]);
      }
      __builtin_amdgcn_wave_barrier();
    }
  }

  // ---- measurement: <Z_q> = sum_i p_i * (1 - 2*bit_{9-q}(i)) ----
  float acc[NQ];
  #pragma unroll
  for (int q = 0; q < NQ; ++q) acc[q] = 0.f;
  #pragma unroll 4
  for (int r = 0; r < 32; ++r) {
    int i = r * 32 + t;
    float2 amp = sv[i];
    float p = amp.x * amp.x + amp.y * amp.y;
    #pragma unroll
    for (int q = 0; q < NQ; ++q)
      acc[q] += ((i >> (9 - q)) & 1) ? -p : p;
  }
  #pragma unroll
  for (int q = 0; q < NQ; ++q) {
    #pragma unroll
    for (int off = 16; off >= 1; off >>= 1)
      acc[q] += __shfl_xor(acc[q], off, 32);
  }
  if (t == 0) {
    #pragma unroll
    for (int q = 0; q < NQ; ++q) out[samp * NQ + q] = acc[q];
  }
}

extern "C" void kernel_launch(void* const* d_in, const int* in_sizes, int n_in,
                              void* d_out, int out_size, void* d_ws,
                              size_t ws_size, hipStream_t stream) {
  const float* inputs = (const float*)d_in[0];   // (8192, 10) f32
  const float* weights = (const float*)d_in[1];  // (80,) f32
  float* out = (float*)d_out;                    // (8192, 10) f32
  float* G = (float*)d_ws;                       // 40 * 16 floats of scratch

  const int B = in_sizes[0] / NQ;                // 8192
  build_gates_kernel<<<dim3(1), dim3(64), 0, stream>>>(weights, G);
  qsim_kernel<<<dim3(B / WPB), dim3(256), 0, stream>>>(inputs, G, out);
}